// GlobalMapEncoder_11940009083661
// MI455X (gfx1250) — compile-verified
//
#include <hip/hip_runtime.h>
#include <hip/hip_bf16.h>

#define B_ 32
#define T_ 15
#define V_ 100
#define D_ 768
#define N_ 160
#define LN_EPS 1e-12f

typedef __attribute__((ext_vector_type(2))) float v2f;
typedef __attribute__((ext_vector_type(8))) float v8f;

// ---------------------------------------------------------------------------
// Kernel 1: per-node gather of gmap_img (visited mean OR candidate segment mean)
// One 256-thread block per (b, n). Each thread owns 3 of the 768 columns.
// ---------------------------------------------------------------------------
__global__ __launch_bounds__(256) void node_gather_kernel(
    const float* __restrict__ emb,       // [B,T,V,D] split_traj_embeds
    const int*   __restrict__ vp_lens,   // [B,T]
    const int*   __restrict__ vpids,     // [B,T]
    const int*   __restrict__ cand_ids,  // [B,T,V]
    float*       __restrict__ gmap_img)  // [B,N,D] (workspace)
{
    const int bn  = blockIdx.x;
    const int b   = bn / N_;
    const int n   = bn % N_;
    const int tid = threadIdx.x;
    float* out_row = gmap_img + (size_t)bn * D_;

    if (n == 0) {  // null/stop node -> zero row
        out_row[tid]       = 0.0f;
        out_row[tid + 256] = 0.0f;
        out_row[tid + 512] = 0.0f;
        return;
    }

    __shared__ int s_len[T_];
    __shared__ int s_visited_t;
    __shared__ int s_nmatch;
    __shared__ int s_list[T_ * V_];   // worst case: every candidate matches

    if (tid == 0) { s_visited_t = -1; s_nmatch = 0; }
    if (tid < T_) s_len[tid] = vp_lens[b * T_ + tid];
    __syncthreads();
    if (tid < T_ && vpids[b * T_ + tid] == n) atomicMax(&s_visited_t, tid);
    __syncthreads();

    float a0 = 0.0f, a1 = 0.0f, a2 = 0.0f;
    float inv;

    if (s_visited_t >= 0) {
        // visited node: masked mean over views at step t*
        const int t   = s_visited_t;
        const int len = s_len[t];
        const float* base = emb + ((size_t)(b * T_ + t)) * V_ * D_;
        for (int v = 0; v < len; ++v) {
            const float* r = base + (size_t)v * D_;
            a0 += r[tid];
            a1 += r[tid + 256];
            a2 += r[tid + 512];
        }
        inv = 1.0f / (float)len;
    } else {
        // candidate segment mean: find all (t,v) with cand_ids==n and v valid
        for (int idx = tid; idx < T_ * V_; idx += 256) {
            const int t = idx / V_;
            const int v = idx - t * V_;
            if (v < s_len[t] && cand_ids[(b * T_ + t) * V_ + v] == n) {
                const int p = atomicAdd(&s_nmatch, 1);
                s_list[p] = idx;
            }
        }
        __syncthreads();
        const int m = s_nmatch;
        const float* base = emb + (size_t)b * T_ * V_ * D_;
        for (int i = 0; i < m; ++i) {
            const float* r = base + (size_t)s_list[i] * D_;
            a0 += r[tid];
            a1 += r[tid + 256];
            a2 += r[tid + 512];
        }
        inv = 1.0f / fmaxf((float)m, 1.0f);
    }

    out_row[tid]       = a0 * inv;
    out_row[tid + 256] = a1 * inv;
    out_row[tid + 512] = a2 * inv;
}

// ---------------------------------------------------------------------------
// Kernel 2: pos matmul with CDNA5 f32 WMMA.
// x = pos_fts @ pos_w + pos_b,  [B*N, 7] @ [7, 768] -> [B*N, 768]
// One wave (32 threads) per 16-row tile; K=7 padded to 8 -> two chained
// V_WMMA_F32_16X16X4_F32 ops per 16x16 output tile; 48 column tiles.
//
// ISA VGPR layouts (CDNA5 ISA):
//   A 16x4 f32 : lane = M (mod 16); lanes 0-15 hold K=0,1 in v0,v1;
//                lanes 16-31 hold K=2,3.
//   B  4x16 f32: lane = N (mod 16); lanes 0-15 hold K rows 0,1 in v0,v1;
//                lanes 16-31 hold K rows 2,3 (mirrors A).
//   C/D 16x16  : lane = N + 16*(M>=8); vgpr r holds row M = r (+8 hi group).
// ---------------------------------------------------------------------------
__global__ __launch_bounds__(32) void pos_wmma_kernel(
    const float* __restrict__ pos_fts,   // [B*N, 7]
    const float* __restrict__ pos_w,     // [7, 768]
    const float* __restrict__ pos_b,     // [768]
    float*       __restrict__ posx)      // [B*N, 768] (workspace, pre-LN x)
{
    const int tile = blockIdx.x;          // 16 rows of B*N per tile
    const int lane = threadIdx.x;         // 0..31
    const int mn   = lane & 15;           // M for A, N for B/C/D
    const bool hi  = lane >= 16;

    // A fragments: row = tile*16 + mn, columns K
    const float* pf = pos_fts + (size_t)(tile * 16 + mn) * 7;
    v2f af0, af1;
    af0.x = hi ? pf[2] : pf[0];
    af0.y = hi ? pf[3] : pf[1];
    af1.x = hi ? pf[6] : pf[4];
    af1.y = hi ? 0.0f  : pf[5];           // K=7 zero pad

    for (int ct = 0; ct < D_ / 16; ++ct) {
        const int c0 = ct * 16;
        const float* w = pos_w + c0 + mn; // column c0+mn, row stride D_
        v2f bf0, bf1;
        bf0.x = hi ? w[2 * D_] : w[0 * D_];
        bf0.y = hi ? w[3 * D_] : w[1 * D_];
        bf1.x = hi ? w[6 * D_] : w[4 * D_];
        bf1.y = hi ? 0.0f      : w[5 * D_];   // K=7 zero pad

        const float pb = pos_b[c0 + mn];
        v8f c;
#pragma unroll
        for (int r = 0; r < 8; ++r) c[r] = pb;   // seed accumulator with bias

        c = __builtin_amdgcn_wmma_f32_16x16x4_f32(
                false, af0, false, bf0, (short)0, c, false, false);
        c = __builtin_amdgcn_wmma_f32_16x16x4_f32(
                false, af1, false, bf1, (short)0, c, false, false);

        const int mbase = tile * 16 + (hi ? 8 : 0);
#pragma unroll
        for (int r = 0; r < 8; ++r)
            posx[(size_t)(mbase + r) * D_ + c0 + mn] = c[r];
    }
}

// ---------------------------------------------------------------------------
// Kernel 3: LayerNorm(posx) + gmap_img + step_emb -> out
// One 256-thread block per (b,n); wave32 shuffle + LDS reduction.
// ---------------------------------------------------------------------------
__global__ __launch_bounds__(256) void finalize_kernel(
    const float* __restrict__ gmap_img,   // [B*N, D]
    const float* __restrict__ posx,       // [B*N, D]
    const float* __restrict__ step_table, // [100, D]
    const int*   __restrict__ step_ids,   // [B*N]
    const float* __restrict__ ln_g,       // [D]
    const float* __restrict__ ln_b,       // [D]
    float*       __restrict__ out)        // [B*N, D]
{
    const int bn  = blockIdx.x;
    const int tid = threadIdx.x;
    const float* xrow = posx + (size_t)bn * D_;

    const float x0 = xrow[tid];
    const float x1 = xrow[tid + 256];
    const float x2 = xrow[tid + 512];

    float s  = x0 + x1 + x2;
    float ss = x0 * x0 + x1 * x1 + x2 * x2;

    // wave32 reduction
#pragma unroll
    for (int off = 16; off > 0; off >>= 1) {
        s  += __shfl_xor(s,  off, 32);
        ss += __shfl_xor(ss, off, 32);
    }
    __shared__ float sm[8], sq[8];
    const int w = tid >> 5, l = tid & 31;
    if (l == 0) { sm[w] = s; sq[w] = ss; }
    __syncthreads();
    if (tid == 0) {
        float ts = 0.0f, tq = 0.0f;
#pragma unroll
        for (int i = 0; i < 8; ++i) { ts += sm[i]; tq += sq[i]; }
        sm[0] = ts; sq[0] = tq;
    }
    __syncthreads();

    const float mean = sm[0] * (1.0f / (float)D_);
    const float var  = sq[0] * (1.0f / (float)D_) - mean * mean;
    const float rstd = rsqrtf(var + LN_EPS);

    const float* se = step_table + (size_t)step_ids[bn] * D_;
    const float* gi = gmap_img + (size_t)bn * D_;
    float* orow = out + (size_t)bn * D_;

    const int   dd[3] = { tid, tid + 256, tid + 512 };
    const float xv[3] = { x0, x1, x2 };
#pragma unroll
    for (int i = 0; i < 3; ++i) {
        const int d = dd[i];
        const float y = (xv[i] - mean) * rstd * ln_g[d] + ln_b[d];
        orow[d] = gi[d] + se[d] + y;
    }
}

// ---------------------------------------------------------------------------
extern "C" void kernel_launch(void* const* d_in, const int* in_sizes, int n_in,
                              void* d_out, int out_size, void* d_ws, size_t ws_size,
                              hipStream_t stream) {
    const float* split_traj_embeds = (const float*)d_in[0];  // [B,T,V,D]
    const float* gmap_pos_fts      = (const float*)d_in[1];  // [B,N,7]
    const float* step_emb_table    = (const float*)d_in[2];  // [100,D]
    const float* pos_w             = (const float*)d_in[3];  // [7,D]
    const float* pos_b             = (const float*)d_in[4];  // [D]
    const float* ln_g              = (const float*)d_in[5];  // [D]
    const float* ln_b              = (const float*)d_in[6];  // [D]
    const int*   split_traj_vp_lens= (const int*)d_in[7];    // [B,T]
    const int*   traj_vpids        = (const int*)d_in[8];    // [B,T]
    const int*   traj_cand_vpids   = (const int*)d_in[9];    // [B,T,V]
    const int*   gmap_step_ids     = (const int*)d_in[10];   // [B,N]
    // d_in[11] gmap_lens is unused by the reference.

    float* out = (float*)d_out;

    float* ws_gmap_img = (float*)d_ws;                        // B*N*D floats
    float* ws_posx     = ws_gmap_img + (size_t)B_ * N_ * D_;  // B*N*D floats

    // Kernel 1: gmap_img (visited / candidate segment mean), node 0 zeroed
    node_gather_kernel<<<B_ * N_, 256, 0, stream>>>(
        split_traj_embeds, split_traj_vp_lens, traj_vpids, traj_cand_vpids,
        ws_gmap_img);

    // Kernel 2: pos matmul via f32 WMMA -> pre-LN x
    pos_wmma_kernel<<<(B_ * N_) / 16, 32, 0, stream>>>(
        gmap_pos_fts, pos_w, pos_b, ws_posx);

    // Kernel 3: LN + sum of the three terms
    finalize_kernel<<<B_ * N_, 256, 0, stream>>>(
        ws_gmap_img, ws_posx, step_emb_table, gmap_step_ids, ln_g, ln_b, out);
}